// GraphNeuralAnomalyDetector_65824668778575
// MI455X (gfx1250) — compile-verified
//
#include <hip/hip_runtime.h>
#include <hip/hip_bf16.h>
#include <math.h>

#define N_NODES 100000
#define N_EDGES 1600000
#define DIM 64
#define ROW_TILES (N_NODES / 16)   // 6250, exact

typedef __attribute__((ext_vector_type(2))) float v2f;
typedef __attribute__((ext_vector_type(8))) float v8f;

// ---------------------------------------------------------------- zero fill
__global__ void zero_kernel(float* __restrict__ p, long long n) {
    long long i = (long long)blockIdx.x * blockDim.x + threadIdx.x;
    long long stride = (long long)gridDim.x * blockDim.x;
    for (; i < n; i += stride) p[i] = 0.0f;
}

// ------------------------------------------------- edge scatter: agg[dst] += w*h[src]
// 16 lanes per edge, float4 per lane (64 floats per edge row)
__global__ __launch_bounds__(256) void scatter_kernel(
        const float* __restrict__ h,
        const int*   __restrict__ edge_index,   // [2, E] row-major: src row then dst row
        const float* __restrict__ ew,
        float* __restrict__ agg) {
    long long gid = (long long)blockIdx.x * blockDim.x + threadIdx.x;
    long long e = gid >> 4;
    if (e >= N_EDGES) return;
    int lane = (int)(gid & 15);
    int src = edge_index[e];
    int dst = edge_index[(long long)N_EDGES + e];
    float w = ew[e];
    float4 v = ((const float4*)(h + (size_t)src * DIM))[lane];
    float* ap = agg + (size_t)dst * DIM + lane * 4;
    atomicAdd(ap + 0, w * v.x);
    atomicAdd(ap + 1, w * v.y);
    atomicAdd(ap + 2, w * v.z);
    atomicAdd(ap + 3, w * v.w);
}

// ------------------------------------------------- OUT[N,64] = AGG[N,64] @ W^T + b
// One wave computes a 16x64 output tile using V_WMMA_F32_16X16X4_F32.
// A frag (16x4): lane<16 holds K=k0,k0+1 ; lane>=16 holds K=k0+2,k0+3  (ISA 7.12.2)
// B frag (4x16): B[k][j] = W[j][k]; same K split across lane halves, N = lane&15
// C/D: VGPR c -> row (c + 8*half), col = lane&15
template <bool RELU>
__global__ __launch_bounds__(256) void gemm_wmma_kernel(
        const float* __restrict__ A,
        const float* __restrict__ W,       // [64,64] row-major, out = A @ W^T
        const float* __restrict__ bias,    // [64]
        float* __restrict__ out) {
    __shared__ float Wlds[DIM * DIM];      // 16 KB of the 320 KB/WGP LDS
    for (int i = threadIdx.x; i < DIM * DIM / 4; i += 256)
        ((float4*)Wlds)[i] = ((const float4*)W)[i];
    __syncthreads();

    const int wave = threadIdx.x >> 5;
    const int tile = blockIdx.x * 8 + wave;
    if (tile >= ROW_TILES) return;         // wave-uniform: EXEC all-1s for WMMA
    const int row0 = tile * 16;
    const int lane = threadIdx.x & 31;
    const int hl   = lane >> 4;            // lane half
    const int m    = lane & 15;
    const float* arow = A + (size_t)(row0 + m) * DIM;

    v8f acc0 = {}, acc1 = {}, acc2 = {}, acc3 = {};
    #pragma unroll
    for (int k0 = 0; k0 < DIM; k0 += 4) {
        const int ka = k0 + 2 * hl;
        v2f a;  a.x = arow[ka];             a.y = arow[ka + 1];
        v2f b0; b0.x = Wlds[(     m) * DIM + ka]; b0.y = Wlds[(     m) * DIM + ka + 1];
        v2f b1; b1.x = Wlds[(16 + m) * DIM + ka]; b1.y = Wlds[(16 + m) * DIM + ka + 1];
        v2f b2; b2.x = Wlds[(32 + m) * DIM + ka]; b2.y = Wlds[(32 + m) * DIM + ka + 1];
        v2f b3; b3.x = Wlds[(48 + m) * DIM + ka]; b3.y = Wlds[(48 + m) * DIM + ka + 1];
        acc0 = __builtin_amdgcn_wmma_f32_16x16x4_f32(false, a, false, b0, (short)0, acc0, false, false);
        acc1 = __builtin_amdgcn_wmma_f32_16x16x4_f32(false, a, false, b1, (short)0, acc1, false, false);
        acc2 = __builtin_amdgcn_wmma_f32_16x16x4_f32(false, a, false, b2, (short)0, acc2, false, false);
        acc3 = __builtin_amdgcn_wmma_f32_16x16x4_f32(false, a, false, b3, (short)0, acc3, false, false);
    }

    const float bb0 = bias[m], bb1 = bias[16 + m], bb2 = bias[32 + m], bb3 = bias[48 + m];
    #pragma unroll
    for (int c = 0; c < 8; ++c) {
        const int r = row0 + c + 8 * hl;
        float* orow = out + (size_t)r * DIM;
        float v;
        v = acc0[c] + bb0; if (RELU) v = fmaxf(v, 0.0f); orow[     m] = v;
        v = acc1[c] + bb1; if (RELU) v = fmaxf(v, 0.0f); orow[16 + m] = v;
        v = acc2[c] + bb2; if (RELU) v = fmaxf(v, 0.0f); orow[32 + m] = v;
        v = acc3[c] + bb3; if (RELU) v = fmaxf(v, 0.0f); orow[48 + m] = v;
    }
}

// ------------------------------------------------- column sums of h3 into pooled[64]
__global__ __launch_bounds__(256) void pool_kernel(const float* __restrict__ h,
                                                   float* __restrict__ pooled) {
    __shared__ float s[256];
    const int tid = threadIdx.x;
    const int col = tid & 63;
    const int rsub = tid >> 6;                   // 0..3
    float sum = 0.0f;
    for (long long r = (long long)blockIdx.x * 4 + rsub; r < N_NODES;
         r += (long long)gridDim.x * 4)
        sum += h[r * DIM + col];
    s[tid] = sum;
    __syncthreads();
    if (tid < 128) s[tid] += s[tid + 128];
    __syncthreads();
    if (tid < 64) atomicAdd(&pooled[col], s[tid] + s[tid + 64]);
}

// ------------------------------------------------- tiny MLP head -> sigmoid score
__global__ void mlp_kernel(const float* __restrict__ pooled,
                           const float* __restrict__ Wp1, const float* __restrict__ bp1,
                           const float* __restrict__ Wp2, const float* __restrict__ bp2,
                           float* __restrict__ score) {
    const int j = threadIdx.x;                   // 0..31 (one wave)
    const float invN = 1.0f / (float)N_NODES;
    float s = bp1[j];
    #pragma unroll
    for (int k = 0; k < DIM; ++k) s += Wp1[j * DIM + k] * (pooled[k] * invN);
    float z = fmaxf(s, 0.0f);
    float p = z * Wp2[j];
    #pragma unroll
    for (int off = 16; off > 0; off >>= 1) p += __shfl_down(p, off, 32);
    if (j == 0) *score = 1.0f / (1.0f + expf(-(p + bp2[0])));
}

// ------------------------------------------------- broadcast score to d_out[0:N]
__global__ void bcast_kernel(const float* __restrict__ score, float* __restrict__ out) {
    int i = blockIdx.x * blockDim.x + threadIdx.x;
    if (i < N_NODES) out[i] = *score;
}

// ================================================================ launch
extern "C" void kernel_launch(void* const* d_in, const int* in_sizes, int n_in,
                              void* d_out, int out_size, void* d_ws, size_t ws_size,
                              hipStream_t stream) {
    const float* x   = (const float*)d_in[0];
    const int*   ei  = (const int*)  d_in[1];
    const float* ew  = (const float*)d_in[2];
    const float* W0  = (const float*)d_in[3];
    const float* b0  = (const float*)d_in[4];
    const float* W1  = (const float*)d_in[5];
    const float* b1  = (const float*)d_in[6];
    const float* W2  = (const float*)d_in[7];
    const float* b2  = (const float*)d_in[8];
    const float* Wp1 = (const float*)d_in[9];
    const float* bp1 = (const float*)d_in[10];
    const float* Wp2 = (const float*)d_in[11];
    const float* bp2 = (const float*)d_in[12];

    float* out_scores = (float*)d_out;               // [N]
    float* out_h      = out_scores + N_NODES;        // [N,64] final h, also inter-layer buffer

    float* agg    = (float*)d_ws;                    // [N,64] scratch (25.6 MB)
    float* pooled = agg + (size_t)N_NODES * DIM;     // [64]
    float* score  = pooled + DIM;                    // [1]

    const dim3 blk(256);
    const long long aggN = (long long)N_NODES * DIM;
    const int zgrid = 4096;
    const int sgrid = (N_EDGES * 16) / 256;          // 100000 blocks
    const int ggrid = (ROW_TILES + 7) / 8;           // 782 blocks, 8 waves each

    // ---- layer 1: agg = scatter(x); h = relu(agg @ W0^T + b0)
    zero_kernel<<<zgrid, blk, 0, stream>>>(agg, aggN);
    scatter_kernel<<<sgrid, blk, 0, stream>>>(x, ei, ew, agg);
    gemm_wmma_kernel<true><<<ggrid, blk, 0, stream>>>(agg, W0, b0, out_h);

    // ---- layer 2
    zero_kernel<<<zgrid, blk, 0, stream>>>(agg, aggN);
    scatter_kernel<<<sgrid, blk, 0, stream>>>(out_h, ei, ew, agg);
    gemm_wmma_kernel<true><<<ggrid, blk, 0, stream>>>(agg, W1, b1, out_h);

    // ---- layer 3 (no relu), final h written straight into d_out
    zero_kernel<<<zgrid, blk, 0, stream>>>(agg, aggN);
    scatter_kernel<<<sgrid, blk, 0, stream>>>(out_h, ei, ew, agg);
    gemm_wmma_kernel<false><<<ggrid, blk, 0, stream>>>(agg, W2, b2, out_h);

    // ---- mean pool + MLP head + broadcast
    zero_kernel<<<1, 128, 0, stream>>>(pooled, DIM + 1);
    pool_kernel<<<256, blk, 0, stream>>>(out_h, pooled);
    mlp_kernel<<<1, 32, 0, stream>>>(pooled, Wp1, bp1, Wp2, bp2, score);
    bcast_kernel<<<(N_NODES + 255) / 256, blk, 0, stream>>>(score, out_scores);
}